// FastWT_78692390797914
// MI455X (gfx1250) — compile-verified
//
#include <hip/hip_runtime.h>
#include <hip/hip_bf16.h>

// Haar level-1 "high" reconstruction: out = x - mean(2x2 block), exact for Haar.
// x: [B=8, C=64, H=512, W=512] float32, contiguous.
//
// Roofline: 1 GB of one-shot traffic @ 23.3 TB/s ~= 43 us; ~0.3 GFLOP total.
// Pure streaming kernel. CDNA5-relevant choice: non-temporal (TH=NT) hints on
// both the 128-bit loads and stores so the single-use 512MB stream does not
// churn the 192MB L2. WMMA is deliberately NOT used: expressing the averaging
// as dense GEMMs would be ~275 TFLOP of f32 work for a 0.25 FLOP/byte problem.

typedef float v4f __attribute__((ext_vector_type(4)));

#define WIDTH 512              // W (floats per row)
#define GROUPS_PER_ROWPAIR 128 // W / 4

__global__ __launch_bounds__(256)
void FastWT_haar_high_kernel(const float* __restrict__ x,
                             float* __restrict__ out,
                             int nUnits)
{
    int u = blockIdx.x * 256 + threadIdx.x;
    if (u >= nUnits) return;

    // unit -> (row-pair, group of 4 columns)
    int rp = u >> 7;                 // u / GROUPS_PER_ROWPAIR
    int g  = u & (GROUPS_PER_ROWPAIR - 1);

    // element offset of the top float4; bottom row is +WIDTH floats (+2048 B,
    // folded into the instruction's immediate offset by the compiler).
    long long base = (long long)rp * (2 * WIDTH) + ((long long)g << 2);

    const v4f a = __builtin_nontemporal_load((const v4f*)(x + base));         // top row
    const v4f b = __builtin_nontemporal_load((const v4f*)(x + base + WIDTH)); // bottom row

    // Two 2x2 blocks per unit: cols {0,1} -> m0, cols {2,3} -> m1.
    float m0 = (a.x + a.y + b.x + b.y) * 0.25f;
    float m1 = (a.z + a.w + b.z + b.w) * 0.25f;

    v4f oa = { a.x - m0, a.y - m0, a.z - m1, a.w - m1 };
    v4f ob = { b.x - m0, b.y - m0, b.z - m1, b.w - m1 };

    __builtin_nontemporal_store(oa, (v4f*)(out + base));
    __builtin_nontemporal_store(ob, (v4f*)(out + base + WIDTH));
}

extern "C" void kernel_launch(void* const* d_in, const int* in_sizes, int n_in,
                              void* d_out, int out_size, void* d_ws, size_t ws_size,
                              hipStream_t stream)
{
    (void)n_in; (void)d_ws; (void)ws_size; (void)in_sizes;

    const float* x = (const float*)d_in[0];
    float* out     = (float*)d_out;

    // Each unit covers 8 elements (two float4s: top+bottom of two 2x2 blocks).
    int nUnits = out_size / 8;           // 16,777,216 for [8,64,512,512]
    int blocks = (nUnits + 255) / 256;   // 65,536 blocks of 8 wave32s

    FastWT_haar_high_kernel<<<blocks, 256, 0, stream>>>(x, out, nUnits);
}